// HypergraphCompressor_46961172414851
// MI455X (gfx1250) — compile-verified
//
#include <hip/hip_runtime.h>

typedef __attribute__((ext_vector_type(2))) float v2f;
typedef __attribute__((ext_vector_type(8))) float v8f;

// workspace layout (float offsets)
static constexpr int XM_OFF = 0;                      // 1024   : xmean
static constexpr int P_OFF  = 1024;                   // 32*1024: xmean partials
static constexpr int H_OFF  = P_OFF + 32 * 1024;      // 7*100*1024 : h (post-ReLU)
static constexpr int MK_OFF = H_OFF + 7 * 100 * 1024; // 7*128  : masks
static constexpr int NM_OFF = MK_OFF + 7 * 128;       // 7*1024 : node means

// ---------------- xmean = mean over S of x : two-stage deterministic ----------
__global__ __launch_bounds__(128) void xmean_partial(const float* __restrict__ x,
                                                     float* __restrict__ ws) {
  const int col = (blockIdx.x << 7) + threadIdx.x;   // 0..1023
  const int r0  = blockIdx.y << 8;                   // 256-row slab
  const float* xp = x + r0 * 1024 + col;
  float s = 0.f;
  for (int r = 0; r < 256; ++r) s += xp[r << 10];
  ws[P_OFF + (blockIdx.y << 10) + col] = s;
}

__global__ __launch_bounds__(128) void xmean_reduce(float* __restrict__ ws) {
  const int col = (blockIdx.x << 7) + threadIdx.x;
  float s = 0.f;
  for (int b = 0; b < 32; ++b) s += ws[P_OFF + (b << 10) + col];
  ws[XM_OFF + col] = s * (1.0f / 8192.0f);
}

// ---------------- h = relu(gather(x) @ W1[i]) via v_wmma_f32_16x16x4_f32 ------
// grid = (16 n-groups of 64 cols, 4 m-groups of 32 rows, 7 experts); 1 wave/block
// Each wave computes a 32x64 tile: 2 M-tiles x 4 N-subtiles = 8 accumulators,
// so each B (W1) load feeds 2 WMMAs -> half the L2 traffic of a 16-row tile.
__global__ __launch_bounds__(32) void gemm1_relu(const float* __restrict__ x,
                                                 const float* __restrict__ W1,
                                                 const float* __restrict__ b1,
                                                 const int* __restrict__ idx,
                                                 float* __restrict__ ws) {
  const int i     = blockIdx.z;       // expert
  const int k     = i + 2;            // nodes per sample
  const int m0    = blockIdx.y << 5;  // sample tile base (0,32,64,96; rows >=100 padded)
  const int n0    = blockIdx.x << 6;  // output-col group base
  const int lane  = threadIdx.x;
  const int l15   = lane & 15;
  const int khalf = lane >> 4;        // K pair select (ISA A/B layout)
  const int mrow0 = m0 + l15;         // row for M-tile 0
  const int mrow1 = m0 + 16 + l15;    // row for M-tile 1
  const float ms0 = (mrow0 < 100) ? 1.0f : 0.0f;   // branchless pad-row zeroing
  const float ms1 = (mrow1 < 100) ? 1.0f : 0.0f;

  // preload this lane's sample indices for both rows; clamp pad rows to row 99
  const int* ip0 = idx + i * 800 + ((mrow0 < 100) ? mrow0 : 99) * 8;
  const int* ip1 = idx + i * 800 + ((mrow1 < 100) ? mrow1 : 99) * 8;
  int rowoff0[8], rowoff1[8];
#pragma unroll
  for (int j = 0; j < 8; ++j) rowoff0[j] = ip0[j] << 10;
#pragma unroll
  for (int j = 0; j < 8; ++j) rowoff1[j] = ip1[j] << 10;

  const float* wbase = W1 + i * 8388608 + n0 + l15;  // W1[i], col base

  v8f c00 = {}, c01 = {}, c02 = {}, c03 = {};   // rows m0..m0+15
  v8f c10 = {}, c11 = {}, c12 = {}, c13 = {};   // rows m0+16..m0+31

  for (int j = 0; j < k; ++j) {
    const float* xrow0 = x + rowoff0[j] + (khalf << 1);
    const float* xrow1 = x + rowoff1[j] + (khalf << 1);
    const float* wrow  = wbase + (((j << 10) + (khalf << 1)) << 10);
    for (int d = 0; d < 1024; d += 4) {
      // A 16x4: lane<16 -> K={d,d+1}, lane>=16 -> K={d+2,d+3}, M = l15
      v2f a0, a1;
      a0.x = xrow0[d]     * ms0;
      a0.y = xrow0[d + 1] * ms0;
      a1.x = xrow1[d]     * ms1;
      a1.y = xrow1[d + 1] * ms1;
      // B 4x16 per 16-col subtile: b.x = W1[row K][col], b.y = next K row
      const float* wp = wrow + (d << 10);
      v2f b0, b1v, b2v, b3v;
      b0.x  = wp[0];   b0.y  = wp[1024];
      b1v.x = wp[16];  b1v.y = wp[1040];
      b2v.x = wp[32];  b2v.y = wp[1056];
      b3v.x = wp[48];  b3v.y = wp[1072];
      c00 = __builtin_amdgcn_wmma_f32_16x16x4_f32(false, a0, false, b0,  (short)0, c00, false, false);
      c10 = __builtin_amdgcn_wmma_f32_16x16x4_f32(false, a1, false, b0,  (short)0, c10, false, false);
      c01 = __builtin_amdgcn_wmma_f32_16x16x4_f32(false, a0, false, b1v, (short)0, c01, false, false);
      c11 = __builtin_amdgcn_wmma_f32_16x16x4_f32(false, a1, false, b1v, (short)0, c11, false, false);
      c02 = __builtin_amdgcn_wmma_f32_16x16x4_f32(false, a0, false, b2v, (short)0, c02, false, false);
      c12 = __builtin_amdgcn_wmma_f32_16x16x4_f32(false, a1, false, b2v, (short)0, c12, false, false);
      c03 = __builtin_amdgcn_wmma_f32_16x16x4_f32(false, a0, false, b3v, (short)0, c03, false, false);
      c13 = __builtin_amdgcn_wmma_f32_16x16x4_f32(false, a1, false, b3v, (short)0, c13, false, false);
    }
  }

  // C/D layout: vgpr v, lane<16 -> (row v, col lane); lane>=16 -> (row v+8, col lane-16)
  float* hbuf = ws + H_OFF;
  const float bb0 = b1[(i << 10) + n0 + l15];
  const float bb1 = b1[(i << 10) + n0 + l15 + 16];
  const float bb2 = b1[(i << 10) + n0 + l15 + 32];
  const float bb3 = b1[(i << 10) + n0 + l15 + 48];

  const int rbase0 = m0 + (khalf << 3);
#pragma unroll
  for (int v = 0; v < 8; ++v) {
    const int r = rbase0 + v;
    if (r < 100) {
      float* hp = hbuf + ((i * 100 + r) << 10) + n0 + l15;
      hp[0]  = fmaxf(c00[v] + bb0, 0.f);
      hp[16] = fmaxf(c01[v] + bb1, 0.f);
      hp[32] = fmaxf(c02[v] + bb2, 0.f);
      hp[48] = fmaxf(c03[v] + bb3, 0.f);
    }
  }
  const int rbase1 = m0 + 16 + (khalf << 3);
#pragma unroll
  for (int v = 0; v < 8; ++v) {
    const int r = rbase1 + v;
    if (r < 100) {
      float* hp = hbuf + ((i * 100 + r) << 10) + n0 + l15;
      hp[0]  = fmaxf(c10[v] + bb0, 0.f);
      hp[16] = fmaxf(c11[v] + bb1, 0.f);
      hp[32] = fmaxf(c12[v] + bb2, 0.f);
      hp[48] = fmaxf(c13[v] + bb3, 0.f);
    }
  }
}

// ---------------- mask[i][n] = (h[i][n] . W2[i] + b2[i] > 0) -----------------
__global__ __launch_bounds__(32) void gemv_mask(const float* __restrict__ W2,
                                                const float* __restrict__ b2,
                                                float* __restrict__ ws) {
  const int n = blockIdx.x, i = blockIdx.y, t = threadIdx.x;
  const float* hp = ws + H_OFF + ((i * 100 + n) << 10);
  const float* wp = W2 + (i << 10);
  float s = 0.f;
  for (int d = t; d < 1024; d += 32) s += hp[d] * wp[d];
#pragma unroll
  for (int off = 16; off > 0; off >>= 1) s += __shfl_down(s, off, 32);
  if (t == 0) {
    const float z = s + b2[i];          // sigmoid(z) > 0.5  <=>  z > 0
    ws[MK_OFF + (i << 7) + n] = (z > 0.f) ? 1.f : 0.f;
  }
}

// ---------------- node_mean[i] = masked mean of gathered feats ---------------
__global__ __launch_bounds__(128) void node_mean_kernel(const float* __restrict__ x,
                                                        const int* __restrict__ idx,
                                                        float* __restrict__ ws) {
  const int i   = blockIdx.y;
  const int k   = i + 2;
  const int col = (blockIdx.x << 7) + threadIdx.x;
  const float* mk = ws + MK_OFF + (i << 7);
  float cnt = 0.f;
  for (int n = 0; n < 100; ++n) cnt += mk[n];
  float acc = 0.f;
  for (int n = 0; n < 100; ++n) {
    if (mk[n] != 0.f) {                      // uniform across block
      const int* ip = idx + i * 800 + n * 8;
      for (int j = 0; j < k; ++j) acc += x[(ip[j] << 10) + col];
    }
  }
  const float scale = 1.f / ((float)k * fmaxf(cnt, 1.f));
  ws[NM_OFF + (i << 10) + col] = acc * scale;
}

// ---------------- out = node_mean@Wc + bc  +  xmean@Wn + bn ------------------
__global__ __launch_bounds__(128) void final_kernel(const float* __restrict__ Wc,
                                                    const float* __restrict__ bc,
                                                    const float* __restrict__ Wn,
                                                    const float* __restrict__ bn,
                                                    const float* __restrict__ ws,
                                                    float* __restrict__ out) {
  const int o  = (blockIdx.x << 7) + threadIdx.x;  // 0..1791
  const int i  = o >> 8;
  const int oo = o & 255;
  const float* xm = ws + XM_OFF;
  const float* nm = ws + NM_OFF + (i << 10);
  const float* wc = Wc + (i << 18) + oo;           // Wc[i][d][oo], stride 256
  const float* wn = Wn + o;                        // Wn[d][o], stride 1792
  float acc = bn[o] + bc[(i << 8) + oo];
  for (int d = 0; d < 1024; ++d) {
    acc += xm[d] * wn[d * 1792];
    acc += nm[d] * wc[d << 8];
  }
  out[o] = acc;
}

extern "C" void kernel_launch(void* const* d_in, const int* in_sizes, int n_in,
                              void* d_out, int out_size, void* d_ws, size_t ws_size,
                              hipStream_t stream) {
  const float* x   = (const float*)d_in[0];
  const float* W1  = (const float*)d_in[1];
  const float* b1  = (const float*)d_in[2];
  const float* W2  = (const float*)d_in[3];
  const float* b2  = (const float*)d_in[4];
  const float* Wc  = (const float*)d_in[5];
  const float* bc  = (const float*)d_in[6];
  const float* Wn  = (const float*)d_in[7];
  const float* bn  = (const float*)d_in[8];
  const int*   idx = (const int*)d_in[9];
  float* ws  = (float*)d_ws;
  float* out = (float*)d_out;
  (void)in_sizes; (void)n_in; (void)out_size; (void)ws_size;

  xmean_partial   <<<dim3(8, 32),   128, 0, stream>>>(x, ws);
  xmean_reduce    <<<dim3(8),       128, 0, stream>>>(ws);
  gemm1_relu      <<<dim3(16, 4, 7), 32, 0, stream>>>(x, W1, b1, idx, ws);
  gemv_mask       <<<dim3(100, 7),   32, 0, stream>>>(W2, b2, ws);
  node_mean_kernel<<<dim3(8, 7),    128, 0, stream>>>(x, idx, ws);
  final_kernel    <<<dim3(14),      128, 0, stream>>>(Wc, bc, Wn, bn, ws, out);
}